// CTCLoss_56298431315981
// MI455X (gfx1250) — compile-verified
//
#include <hip/hip_runtime.h>

// Problem constants (fixed by the harness's setup_inputs)
constexpr int T_   = 1024;
constexpr int B_   = 32;
constexpr int C_   = 1024;
constexpr int L_   = 128;
constexpr int SP_  = 264;   // padded row stride (floats) for Eext, 16B-aligned

typedef __attribute__((ext_vector_type(2))) float v2f;
typedef __attribute__((ext_vector_type(4))) float v4f;
typedef __attribute__((ext_vector_type(8))) float v8f;

// Exact fp32 sum of one value per lane across a full wave32, using the CDNA5
// matrix pipe: A(16x4) holds p in K=0 (lanes 0-15) and K=2 (lanes 16-31),
// zeros elsewhere; B(4x16)=ones  =>  D[m,n] = p[m] + p[m+16].
// Each lane sums its 8 D registers, then one shfl_xor(16) completes the sum.
// REQUIRES: EXEC all ones at the call site.
__device__ __forceinline__ float wave_sum32_wmma(float p) {
    v2f a; a.x = p;    a.y = 0.0f;
    v2f b; b.x = 1.0f; b.y = 1.0f;
    v8f c = {};
    v8f r = __builtin_amdgcn_wmma_f32_16x16x4_f32(
        /*neg_a=*/false, a, /*neg_b=*/false, b,
        /*c_mod=*/(short)0, c, /*reuse_a=*/false, /*reuse_b=*/false);
    float s = r[0] + r[1] + r[2] + r[3] + r[4] + r[5] + r[6] + r[7];
    s += __shfl_xor(s, 16, 32);
    return s;
}

// ---------------------------------------------------------------------------
// Kernel 1: one wave per (t,b) row. Computes row max + sum(exp) (softmax
// denominator) and writes the gathered, normalized linear-space emissions
// Eext[row][s] = softmax(data[row])[ext_class(s)],  s = 0..256.
// Even s -> blank (class 0); odd s -> labels[b][(s-1)/2].
// Bulk row reads are non-temporal so the 128 MB logit stream does not evict
// the ~33 MB Eext working set from the 192 MB L2 before the DP phase.
// ---------------------------------------------------------------------------
__global__ void ctc_rowprep(const float* __restrict__ data,
                            const int*   __restrict__ labels,
                            float*       __restrict__ Eext) {
    const int row  = blockIdx.x * 8 + (threadIdx.x >> 5);   // t*B + b
    const int lane = threadIdx.x & 31;
    const int b    = row & (B_ - 1);

    const float* dr  = data + (size_t)row * C_;
    const v4f*   dp4 = (const v4f*)dr;

    v4f v[8];
#pragma unroll
    for (int c = 0; c < 8; ++c) v[c] = __builtin_nontemporal_load(dp4 + c * 32 + lane);

    // row max
    float m = v[0].x;
#pragma unroll
    for (int c = 0; c < 8; ++c) {
        m = fmaxf(m, v[c].x); m = fmaxf(m, v[c].y);
        m = fmaxf(m, v[c].z); m = fmaxf(m, v[c].w);
    }
#pragma unroll
    for (int off = 16; off >= 1; off >>= 1) m = fmaxf(m, __shfl_xor(m, off, 32));

    // sum of exp(x - m): per-lane partial, then WMMA wave reduction
    float p = 0.0f;
#pragma unroll
    for (int c = 0; c < 8; ++c) {
        p += __expf(v[c].x - m); p += __expf(v[c].y - m);
        p += __expf(v[c].z - m); p += __expf(v[c].w - m);
    }
    const float total = wave_sum32_wmma(p);          // EXEC all ones here
    const float rinv  = __builtin_amdgcn_rcpf(total);

    // blank emission = softmax prob of class 0 (held by lane 0, v[0].x)
    const float blankRaw = __shfl(v[0].x, 0, 32);
    const float blankE   = __expf(blankRaw - m) * rinv;

    const int* lb = labels + b * L_;
    float*     eo = Eext + (size_t)row * SP_;

#pragma unroll
    for (int k = 0; k < 8; ++k) {
        const int s = k * 32 + lane;
        float e;
        if (s & 1) {
            const int cls = lb[(s - 1) >> 1];
            e = __expf(dr[cls] - m) * rinv;          // row is hot in cache
        } else {
            e = blankE;
        }
        eo[s] = e;
    }
    if (lane == 0) eo[256] = blankE;                 // state 256 (blank)
}

// ---------------------------------------------------------------------------
// Kernel 2: CTC forward DP, one wave32 per example b. Linear-space alpha with
// max-rescaling every 4 steps (log-scale carried separately). Lane holds the
// 8 contiguous states s = lane*8 .. lane*8+7; lane 31 also owns state 256.
// Cross-lane traffic per step: two __shfl_up's. Emission loads are software-
// pipelined one step ahead (registers) + global_prefetch_b8 at distance 12.
// ---------------------------------------------------------------------------
__global__ void ctc_dp(const float* __restrict__ Eext,
                       const int*   __restrict__ labels,
                       const int*   __restrict__ label_length,
                       const int*   __restrict__ data_length,
                       float*       __restrict__ lossB) {
    const int b    = blockIdx.x;
    const int lane = threadIdx.x;
    const int* lb  = labels + b * L_;
    const int Tlen = data_length[b];
    const int llen = label_length[b];

    // skip-transition legality for my 8 states (constant over t)
    float sk[8];
#pragma unroll
    for (int j = 0; j < 8; ++j) {
        const int s = lane * 8 + j;
        float allow = 0.0f;
        if (s & 1) {
            const int l   = (s - 1) >> 1;
            const int cur = lb[l];
            const int prv = (l > 0) ? lb[l - 1] : -1;
            if (cur != 0 && cur != prv) allow = 1.0f;
        }
        sk[j] = allow;
    }

    const float* r0 = Eext + (size_t)b * SP_;        // row (t=0, b)
    float a0 = 0.f, a1 = 0.f, a2 = 0.f, a3 = 0.f, a4 = 0.f,
          a5 = 0.f, a6 = 0.f, a7 = 0.f, a8 = 0.f;
    if (lane == 0) { a0 = r0[0]; a1 = r0[1]; }       // alpha_0: only s=0,1 live
    float logscale = 0.0f;

    // software pipeline: preload emissions for t = 1
    const float* rp1 = Eext + (size_t)(1 * B_ + b) * SP_;
    v4f   ce0 = *(const v4f*)(rp1 + lane * 8);
    v4f   ce1 = *(const v4f*)(rp1 + lane * 8 + 4);
    float ceB = rp1[256];

    for (int t = 1; t < Tlen; ++t) {
        // issue next step's loads before computing this step
        const int    tn = (t + 1 < T_) ? (t + 1) : t;
        const float* rn = Eext + (size_t)(tn * B_ + b) * SP_;
        const v4f   ne0 = *(const v4f*)(rn + lane * 8);
        const v4f   ne1 = *(const v4f*)(rn + lane * 8 + 4);
        const float neB = rn[256];
        if (t + 12 < T_)
            __builtin_prefetch(Eext + (size_t)((t + 12) * B_ + b) * SP_, 0, 3);

        float n1 = __shfl_up(a7, 1, 32); if (lane == 0) n1 = 0.0f;  // alpha[s-1] from prev lane
        float n2 = __shfl_up(a6, 1, 32); if (lane == 0) n2 = 0.0f;  // alpha[s-2] from prev lane

        const float t0 = (a0 + n1 + sk[0] * n2) * ce0.x;
        const float t1 = (a1 + a0 + sk[1] * n1) * ce0.y;
        const float t2 = (a2 + a1 + sk[2] * a0) * ce0.z;
        const float t3 = (a3 + a2 + sk[3] * a1) * ce0.w;
        const float t4 = (a4 + a3 + sk[4] * a2) * ce1.x;
        const float t5 = (a5 + a4 + sk[5] * a3) * ce1.y;
        const float t6 = (a6 + a5 + sk[6] * a4) * ce1.z;
        const float t7 = (a7 + a6 + sk[7] * a5) * ce1.w;
        const float t8 = (lane == 31) ? (a8 + a7) * ceB : 0.0f;     // s=256 (even: no skip)

        a0 = t0; a1 = t1; a2 = t2; a3 = t3; a4 = t4;
        a5 = t5; a6 = t6; a7 = t7; a8 = t8;

        if ((t & 3) == 0) {                          // periodic rescale
            float mx = fmaxf(fmaxf(fmaxf(a0, a1), fmaxf(a2, a3)),
                             fmaxf(fmaxf(a4, a5), fmaxf(a6, a7)));
            mx = fmaxf(mx, a8);
#pragma unroll
            for (int off = 16; off >= 1; off >>= 1) mx = fmaxf(mx, __shfl_xor(mx, off, 32));
            const float inv = __builtin_amdgcn_rcpf(mx);
            logscale += __logf(mx);
            a0 *= inv; a1 *= inv; a2 *= inv; a3 *= inv; a4 *= inv;
            a5 *= inv; a6 *= inv; a7 *= inv; a8 *= inv;
        }

        ce0 = ne0; ce1 = ne1; ceB = neB;             // rotate pipeline
    }

    __shared__ float sa[SP_];
    sa[lane * 8 + 0] = a0; sa[lane * 8 + 1] = a1; sa[lane * 8 + 2] = a2;
    sa[lane * 8 + 3] = a3; sa[lane * 8 + 4] = a4; sa[lane * 8 + 5] = a5;
    sa[lane * 8 + 6] = a6; sa[lane * 8 + 7] = a7;
    if (lane == 31) sa[256] = a8;
    __syncthreads();

    if (lane == 0) {
        const int si = 2 * llen;
        const float v = sa[si] + sa[si - 1];         // final blank + final label
        lossB[b] = -(__logf(v) + logscale);
    }
}

// ---------------------------------------------------------------------------
// Kernel 3: mean of the 32 per-example losses (WMMA wave-sum again).
// ---------------------------------------------------------------------------
__global__ void ctc_mean(const float* __restrict__ lossB, float* __restrict__ out) {
    const float v = lossB[threadIdx.x];
    const float s = wave_sum32_wmma(v);              // full wave active
    if (threadIdx.x == 0) out[0] = s * (1.0f / (float)B_);
}

extern "C" void kernel_launch(void* const* d_in, const int* in_sizes, int n_in,
                              void* d_out, int out_size, void* d_ws, size_t ws_size,
                              hipStream_t stream) {
    const int*   labels       = (const int*)d_in[0];
    const float* data         = (const float*)d_in[1];
    const int*   label_length = (const int*)d_in[2];
    const int*   data_length  = (const int*)d_in[3];
    float*       out          = (float*)d_out;

    float* Eext  = (float*)d_ws;                     // T*B*SP floats (~33 MB, L2-resident)
    float* lossB = Eext + (size_t)T_ * B_ * SP_;     // 32 floats

    ctc_rowprep<<<(T_ * B_) / 8, 256, 0, stream>>>(data, labels, Eext);
    ctc_dp<<<B_, 32, 0, stream>>>(Eext, labels, label_length, data_length, lossB);
    ctc_mean<<<1, 32, 0, stream>>>(lossB, out);
}